// Model_7902739824782
// MI455X (gfx1250) — compile-verified
//
#include <hip/hip_runtime.h>
#include <hip/hip_bf16.h>
#include <math.h>

// ---------------------------------------------------------------------------
// Problem constants (from reference): B=32, Tds=2048, L=512, D=256, C=D/4=64
// ---------------------------------------------------------------------------
#define BATCH   32
#define TDS     2048
#define LTXT    512
#define DMODEL  256
#define CDIM    64
#define RATIO   4

// LDS row paddings chosen so row stride % 64 banks = 4 -> conflict-free frag reads
#define QPAD    260     // q tile row stride (floats)
#define SPAD    516     // score tile row stride (floats)

typedef __attribute__((ext_vector_type(2))) float v2f;
typedef __attribute__((ext_vector_type(8))) float v8f;

// ---------------------------------------------------------------------------
// Kernel 1: k_pe[b,l,d] = k[b,l,d] + PE(l, d, w_s[b]),
//           w_s[b] = (mel_len[b]/RATIO) / text_len[b]  (float math, like ref)
// k is read once (NT load); k_pe is re-read 128x per batch -> default RT store
// so it stays L2-resident (16 MB of 192 MB).
// ---------------------------------------------------------------------------
__global__ __launch_bounds__(256) void kpe_kernel(const float* __restrict__ k,
                                                  const int* __restrict__ text_len,
                                                  const int* __restrict__ mel_len,
                                                  float* __restrict__ kpe) {
    int i = blockIdx.x * 256 + threadIdx.x;      // over B*L*D = 4,194,304
    int d = i & (DMODEL - 1);
    int l = (i >> 8) & (LTXT - 1);
    int b = i >> 17;
    float w   = ((float)mel_len[b] / (float)RATIO) / (float)text_len[b];
    int pair  = d >> 1;
    float div = powf(10000.0f, (float)(2 * pair) * (1.0f / (float)DMODEL));
    float ang = w * (float)l / div;
    float pe  = (d & 1) ? cosf(ang) : sinf(ang);
    kpe[i] = __builtin_nontemporal_load(&k[i]) + pe;
}

// ---------------------------------------------------------------------------
// Kernel 2: one workgroup (8 wave32) per (b, 16-row tile of Tds).
//   Phase 0: q tile + PE(w=1), scaled by D^-0.5, into LDS (NT loads of q)
//   Phase 1: scores(16x512) = q_tile @ k_pe^T via v_wmma_f32_16x16x4_f32
//   Phase 2: masked softmax over L, argmax (first occurrence), NT align store
//   Phase 3: out tile (16 t x 64 c) = gather(v, idx) @ w_compress^T via WMMA
// ---------------------------------------------------------------------------
__global__ __launch_bounds__(256) void attn_kernel(const float* __restrict__ q,
                                                   const float* __restrict__ kpe,
                                                   const float* __restrict__ v,
                                                   const float* __restrict__ wc,
                                                   const int* __restrict__ text_len,
                                                   const int* __restrict__ mel_len,
                                                   float* __restrict__ out,
                                                   float* __restrict__ align) {
    __shared__ float qs[16 * QPAD];   // 16.25 KB
    __shared__ float sc[16 * SPAD];   // 33 KB
    __shared__ int   idx_sh[16];

    const int wg   = blockIdx.x;
    const int b    = wg >> 7;                  // Tds/16 = 128 tiles per batch
    const int t0   = (wg & 127) * 16;
    const int tid  = threadIdx.x;
    const int lane = tid & 31;
    const int wave = tid >> 5;
    const int m    = lane & 15;                // A-frag row / B,C-frag column id
    const int hi   = lane >> 4;                // half-wave: selects K pair / M+8

    const int tlen    = text_len[b];
    const int mlen_ds = mel_len[b] / RATIO;

    // ---- Phase 0: q tile + positional encoding (w=1), scaled -------------
    const float inv_sqrt_d = 0.0625f;          // 256^-0.5
    for (int i = tid; i < 16 * DMODEL; i += 256) {
        int mr = i >> 8;                       // row in tile
        int d  = i & (DMODEL - 1);
        int pair  = d >> 1;
        float div = powf(10000.0f, (float)(2 * pair) * (1.0f / (float)DMODEL));
        float ang = (float)(t0 + mr) / div;
        float pe  = (d & 1) ? cosf(ang) : sinf(ang);
        float qv  = __builtin_nontemporal_load(
                        &q[((size_t)b * TDS + t0 + mr) * DMODEL + d]);
        qs[mr * QPAD + d] = (qv + pe) * inv_sqrt_d;
    }
    __syncthreads();

    // ---- Phase 1: scores = q_tile @ kpe^T --------------------------------
    // f32 WMMA fragment layout (ISA 7.12.2):
    //   A 16x4 : lane m holds M=m; VGPR0=K(2*hi), VGPR1=K(2*hi+1)
    //   B 4x16 : lane holds N=lane&15; VGPR0=K(2*hi), VGPR1=K(2*hi+1)
    //   C 16x16: VGPR r -> M = r + 8*hi, N = lane&15
    const float* kb = kpe + (size_t)b * LTXT * DMODEL;
    for (int nt = 0; nt < 4; ++nt) {
        const int n0 = wave * 64 + nt * 16;
        const float* krow = kb + (size_t)(n0 + m) * DMODEL;  // B col = k row l
        v8f c = {};
        #pragma unroll 8
        for (int k0 = 0; k0 < DMODEL; k0 += 4) {
            v2f a  = *(const v2f*)(&qs[m * QPAD + k0 + 2 * hi]);
            v2f bm = *(const v2f*)(krow + k0 + 2 * hi);
            c = __builtin_amdgcn_wmma_f32_16x16x4_f32(
                    false, a, false, bm, (short)0, c, false, false);
        }
        #pragma unroll
        for (int r = 0; r < 8; ++r)
            sc[(r + 8 * hi) * SPAD + n0 + m] = c[r];
    }
    __syncthreads();

    // ---- Phase 2: masked softmax + argmax, 2 rows per wave ---------------
    for (int rr = 0; rr < 2; ++rr) {
        const int row = wave * 2 + rr;
        const int t   = t0 + row;
        const bool mel_ok = (t < mlen_ds);

        float vals[16];
        float vmax = -INFINITY;
        int   vidx = LTXT;                     // first-occurrence argmax
        #pragma unroll
        for (int j = 0; j < 16; ++j) {
            int   l = lane + 32 * j;
            float x = (l < tlen) ? sc[row * SPAD + l] : -INFINITY;
            vals[j] = x;
            if (x > vmax) { vmax = x; vidx = l; }
        }
        #pragma unroll
        for (int off = 16; off > 0; off >>= 1) {
            float om = __shfl_xor(vmax, off, 32);
            int   oi = __shfl_xor(vidx, off, 32);
            if (om > vmax || (om == vmax && oi < vidx)) { vmax = om; vidx = oi; }
        }
        float sum = 0.0f;
        #pragma unroll
        for (int j = 0; j < 16; ++j) {
            float e = __expf(vals[j] - vmax);  // exp(-inf)=0 for masked cols
            vals[j] = e;
            sum += e;
        }
        #pragma unroll
        for (int off = 16; off > 0; off >>= 1) sum += __shfl_xor(sum, off, 32);

        const float scale = mel_ok ? (1.0f / sum) : 0.0f;
        float* arow = align + ((size_t)b * TDS + t) * LTXT;
        #pragma unroll
        for (int j = 0; j < 16; ++j)           // coalesced 128B NT stores
            __builtin_nontemporal_store(vals[j] * scale, &arow[lane + 32 * j]);

        if (lane == 0) idx_sh[row] = mel_ok ? vidx : -1;
    }
    __syncthreads();

    // ---- Phase 3: out tile = gather(v, idx) @ w_compress^T (waves 0..3) --
    if (wave < 4) {
        const int n0 = wave * 16;              // c tile
        const float* wrow = wc + (size_t)(n0 + m) * DMODEL;
        const int src  = idx_sh[m];
        const float flag = (src >= 0) ? 1.0f : 0.0f;     // zero mel-masked rows
        const int srcc = (src >= 0) ? src : 0;           // keep loads uniform
        const float* vrow = v + ((size_t)b * LTXT + srcc) * DMODEL;

        v8f c = {};
        #pragma unroll 8
        for (int k0 = 0; k0 < DMODEL; k0 += 4) {
            v2f a = *(const v2f*)(vrow + k0 + 2 * hi);
            a.x *= flag; a.y *= flag;
            v2f bm = *(const v2f*)(wrow + k0 + 2 * hi);
            c = __builtin_amdgcn_wmma_f32_16x16x4_f32(
                    false, a, false, bm, (short)0, c, false, false);
        }
        // C: lane owns column c = n0+m, rows t = t0 + r + 8*hi -> 8 contiguous t
        float* obase = out + ((size_t)b * CDIM + n0 + m) * TDS + t0 + 8 * hi;
        #pragma unroll
        for (int r = 0; r < 8; ++r)
            __builtin_nontemporal_store(c[r], &obase[r]);
    }
}

// ---------------------------------------------------------------------------
// Inputs (setup_inputs order): q, k, v, w_compress, text_lengths, mel_lengths
// Output: concat( out (B*64*2048 f32), align (B*2048*512 f32) )
// Workspace: k_pe, B*L*D floats = 16 MB
// ---------------------------------------------------------------------------
extern "C" void kernel_launch(void* const* d_in, const int* in_sizes, int n_in,
                              void* d_out, int out_size, void* d_ws, size_t ws_size,
                              hipStream_t stream) {
    const float* q  = (const float*)d_in[0];
    const float* k  = (const float*)d_in[1];
    const float* v  = (const float*)d_in[2];
    const float* wc = (const float*)d_in[3];
    const int* tl   = (const int*)d_in[4];
    const int* ml   = (const int*)d_in[5];

    float* out   = (float*)d_out;                      // (B, 64, Tds)
    float* align = out + (size_t)BATCH * CDIM * TDS;   // (B, Tds, L)
    float* kpe   = (float*)d_ws;                       // (B, L, D)

    const int n_kpe = BATCH * LTXT * DMODEL;
    kpe_kernel<<<n_kpe / 256, 256, 0, stream>>>(k, tl, ml, kpe);

    const int n_wg = BATCH * (TDS / 16);               // 4096 workgroups
    attn_kernel<<<n_wg, 256, 0, stream>>>(q, kpe, v, wc, tl, ml, out, align);
}